// HEPT_48464410968554
// MI455X (gfx1250) — compile-verified
//
#include <hip/hip_runtime.h>

#define N_HASHES 3
#define N_HEADS 8
#define PADDED 32768
#define BLK 128
#define DV 24
#define DQK 27
#define DPAD 28
#define NB (PADDED / BLK) /* 256 */

typedef __attribute__((ext_vector_type(2))) float v2f;
typedef __attribute__((ext_vector_type(8))) float v8f;

// ---- monotone float <-> uint mapping (order-preserving, for sort/atomics) ----
__device__ __forceinline__ unsigned f2mono(float f) {
  unsigned u = __float_as_uint(f);
  return (u & 0x80000000u) ? ~u : (u | 0x80000000u);
}
__device__ __forceinline__ float mono2f(unsigned u) {
  u = (u & 0x80000000u) ? (u & 0x7FFFFFFFu) : ~u;
  return __uint_as_float(u);
}

// ---------------------------------------------------------------------------
// K0: init the per-(r,h) max/min slots
// ---------------------------------------------------------------------------
__global__ void hept_init(unsigned* __restrict__ mm) {
  int t = threadIdx.x;
  if (t < N_HASHES * N_HEADS) {
    mm[2 * t]     = 0u;           // running max (monotone encoding)
    mm[2 * t + 1] = 0xFFFFFFFFu;  // running min
  }
}

// ---------------------------------------------------------------------------
// K1: hash projection  q_hashed/k_hashed = einsum('hnd,hdr->rhn') + global
//     max/min reduction (atomic on monotone-encoded uints)
// grid: (N/256, H, R), block 256
// ---------------------------------------------------------------------------
__global__ __launch_bounds__(256)
void hept_hash(const float* __restrict__ q, const float* __restrict__ k,
               const float* __restrict__ alpha,
               float* __restrict__ qh, float* __restrict__ kh,
               unsigned* __restrict__ mm) {
  const int tid = threadIdx.x;
  const int n = blockIdx.x * 256 + tid;
  const int h = blockIdx.y, r = blockIdx.z;
  const size_t base = ((size_t)h * PADDED + n) * DQK;
  float qd = 0.f, kd = 0.f;
#pragma unroll
  for (int d = 0; d < DQK; ++d) {
    float a = alpha[(h * DQK + d) * N_HASHES + r];  // block-uniform -> SMEM path
    qd = fmaf(q[base + d], a, qd);
    kd = fmaf(k[base + d], a, kd);
  }
  const size_t ob = ((size_t)r * N_HEADS + h) * PADDED + n;
  qh[ob] = qd;
  kh[ob] = kd;

  __shared__ float smx[256], smn[256];
  smx[tid] = fmaxf(qd, kd);
  smn[tid] = fminf(qd, kd);
  __syncthreads();
  for (int s = 128; s > 0; s >>= 1) {
    if (tid < s) {
      smx[tid] = fmaxf(smx[tid], smx[tid + s]);
      smn[tid] = fminf(smn[tid], smn[tid + s]);
    }
    __syncthreads();
  }
  if (tid == 0) {
    unsigned* slot = mm + 2 * (r * N_HEADS + h);
    atomicMax(slot, f2mono(smx[0]));
    atomicMin(slot + 1, f2mono(smn[0]));
  }
}

// ---------------------------------------------------------------------------
// K2: full 32768-element argsort per (r,h,{q|k}) — bitonic sort entirely in
//     LDS (32768 x u64 = 256KB of the 320KB WGP LDS). key = monotone(f)<<32|idx
// grid: 48 blocks of 1024 threads, 262144 B dynamic LDS
// ---------------------------------------------------------------------------
__global__ __launch_bounds__(1024)
void hept_sort(const float* __restrict__ qh, const float* __restrict__ kh,
               const int* __restrict__ shifts, const unsigned* __restrict__ mm,
               int* __restrict__ qpos, int* __restrict__ kpos) {
  extern __shared__ unsigned long long sm[];
  const int bid = blockIdx.x;
  const int which = bid & 1;   // 0 = q, 1 = k
  const int rh = bid >> 1;     // r*H + h
  const size_t base = (size_t)rh * PADDED;
  const float shift = mono2f(mm[2 * rh]) - mono2f(mm[2 * rh + 1]);
  const float* src = which ? kh : qh;
  const int tid = threadIdx.x;

  for (int i = tid; i < PADDED; i += 1024) {
    float f = src[base + i] + (float)shifts[base + i] * shift;
    sm[i] = ((unsigned long long)f2mono(f) << 32) | (unsigned)i;
  }
  __syncthreads();

  for (int kk = 2; kk <= PADDED; kk <<= 1) {
    for (int j = kk >> 1; j > 0; j >>= 1) {
      for (int i = tid; i < PADDED; i += 1024) {
        int ixj = i ^ j;
        if (ixj > i) {
          unsigned long long a = sm[i], b = sm[ixj];
          bool up = ((i & kk) == 0);
          if ((a > b) == up) { sm[i] = b; sm[ixj] = a; }
        }
      }
      __syncthreads();
    }
  }
  int* dst = which ? kpos : qpos;
  for (int i = tid; i < PADDED; i += 1024)
    dst[base + i] = (int)(unsigned)(sm[i] & 0xFFFFFFFFull);
}

// ---------------------------------------------------------------------------
// K3: gather + blocked RBF attention using V_WMMA_F32_16X16X4_F32
// grid: (nb=256, H=8, R=3), block 256 (8 wave32; wave w owns query rows 16w..)
// ---------------------------------------------------------------------------
__global__ __launch_bounds__(256)
void hept_attn(const float* __restrict__ query, const float* __restrict__ key,
               const float* __restrict__ value,
               const int* __restrict__ qpos, const int* __restrict__ kpos,
               float* __restrict__ out) {
  __shared__ float Qs[BLK][DPAD];
  __shared__ float Ks[BLK][DPAD];
  __shared__ float Vs[BLK][32];      // cols 24..31 zero
  __shared__ float qsq[BLK], ksq[BLK];
  __shared__ float stile[8][16][17]; // per-wave S staging (D-layout -> A-layout)

  const int tid = threadIdx.x;
  const int g = blockIdx.x, h = blockIdx.y, r = blockIdx.z;
  const int rh = r * N_HEADS + h;
  const int* qp = qpos + (size_t)rh * PADDED + g * BLK;
  const int* kp = kpos + (size_t)rh * PADDED + g * BLK;

  // ---- gather sorted Q/K/V rows into LDS (2 threads per row) ----
  {
    const int row = tid >> 1;
    const int half = tid & 1;
    const int qi = qp[row], ki = kp[row];
    const float* qsrc = query + ((size_t)h * PADDED + qi) * DQK;
    const float* ksrc = key + ((size_t)h * PADDED + ki) * DQK;
#pragma unroll
    for (int ii = 0; ii < 14; ++ii) {
      int d = half * 14 + ii;
      Qs[row][d] = (d < DQK) ? qsrc[d] : 0.f;
      Ks[row][d] = (d < DQK) ? ksrc[d] : 0.f;
    }
    const float* vsrc = value + ((size_t)h * PADDED + ki) * DV;
#pragma unroll
    for (int ii = 0; ii < 16; ++ii) {
      int d = half * 16 + ii;
      Vs[row][d] = (d < DV) ? vsrc[d] : 0.f;
    }
  }
  __syncthreads();

  // ---- per-row squared norms ----
  if (tid < BLK) {
    float s = 0.f;
#pragma unroll
    for (int d = 0; d < DQK; ++d) s = fmaf(Qs[tid][d], Qs[tid][d], s);
    qsq[tid] = s;
  } else {
    const int t = tid - BLK;
    float s = 0.f;
#pragma unroll
    for (int d = 0; d < DQK; ++d) s = fmaf(Ks[t][d], Ks[t][d], s);
    ksq[t] = s;
  }
  __syncthreads();

  const int wave = tid >> 5;   // query tile (rows 16*wave .. +15)
  const int lane = tid & 31;
  const int lm = lane & 15;    // lane % 16
  const int lh = lane >> 4;    // lane / 16
  const int k0 = 2 * lh;       // A/B fragment K offset for f32 16x16x4

  // preload Q A-fragments: A[m][k], m = lm, k = 4*kk + k0 (+1)
  v2f aq[7];
#pragma unroll
  for (int kk = 0; kk < 7; ++kk) {
    aq[kk].x = Qs[wave * 16 + lm][4 * kk + k0];
    aq[kk].y = Qs[wave * 16 + lm][4 * kk + k0 + 1];
  }
  float qv[8];  // qsq in D-layout order: m = vg + 8*lh
#pragma unroll
  for (int vg = 0; vg < 8; ++vg) qv[vg] = qsq[wave * 16 + vg + 8 * lh];

  v8f oacc0 = {}, oacc1 = {};

  for (int j = 0; j < 8; ++j) {
    // ---- S tile: D = Q * K^T over Dpad=28 (7 x K4 WMMAs) ----
    v8f acc = {};
#pragma unroll
    for (int kk = 0; kk < 7; ++kk) {
      v2f b;  // B[k][n] = Ks[n][k], n = lm
      b.x = Ks[j * 16 + lm][4 * kk + k0];
      b.y = Ks[j * 16 + lm][4 * kk + k0 + 1];
      acc = __builtin_amdgcn_wmma_f32_16x16x4_f32(
          false, aq[kk], false, b, (short)0, acc, false, false);
    }
    // ---- S = exp(min(dot - 0.5*(|q|^2+|k|^2), 0)); stage to LDS ----
    const float kv = ksq[j * 16 + lm];
#pragma unroll
    for (int vg = 0; vg < 8; ++vg) {
      float val = acc[vg] - 0.5f * (qv[vg] + kv);
      stile[wave][vg + 8 * lh][lm] = __expf(fminf(val, 0.f));
    }
    __builtin_amdgcn_wave_barrier();  // DS in-order per wave; keep sched order

    // ---- O += S * V : A = S (16x16), B = V tile rows j*16.. (16 x 24) ----
    v2f sa[4];
#pragma unroll
    for (int kk = 0; kk < 4; ++kk) {
      sa[kk].x = stile[wave][lm][4 * kk + k0];
      sa[kk].y = stile[wave][lm][4 * kk + k0 + 1];
    }
#pragma unroll
    for (int kk = 0; kk < 4; ++kk) {
      const int krow = j * 16 + 4 * kk + k0;
      v2f vb0, vb1;
      vb0.x = Vs[krow][lm];        vb0.y = Vs[krow + 1][lm];
      vb1.x = Vs[krow][16 + lm];   vb1.y = Vs[krow + 1][16 + lm];
      oacc0 = __builtin_amdgcn_wmma_f32_16x16x4_f32(
          false, sa[kk], false, vb0, (short)0, oacc0, false, false);
      oacc1 = __builtin_amdgcn_wmma_f32_16x16x4_f32(
          false, sa[kk], false, vb1, (short)0, oacc1, false, false);
    }
    __builtin_amdgcn_wave_barrier();
  }

  // ---- store O (R,H,nb,B,Dv); D-layout: m = vg + 8*lh, n = lm (+16) ----
  const size_t ob = (((size_t)rh * NB + g) * BLK) * DV;
#pragma unroll
  for (int vg = 0; vg < 8; ++vg) {
    const int m = wave * 16 + vg + 8 * lh;
    out[ob + (size_t)m * DV + lm] = oacc0[vg];
    if (lm < 8) out[ob + (size_t)m * DV + 16 + lm] = oacc1[vg];
  }
}

// ---------------------------------------------------------------------------
extern "C" void kernel_launch(void* const* d_in, const int* in_sizes, int n_in,
                              void* d_out, int out_size, void* d_ws, size_t ws_size,
                              hipStream_t stream) {
  const float* query  = (const float*)d_in[0];
  const float* key    = (const float*)d_in[1];
  const float* value  = (const float*)d_in[2];
  const int*   shifts = (const int*)d_in[3];
  const float* alpha  = (const float*)d_in[4];
  float* out = (float*)d_out;

  const size_t RHN = (size_t)N_HASHES * N_HEADS * PADDED;  // 786432
  float* qh = (float*)d_ws;
  float* kh = qh + RHN;
  unsigned* mm = (unsigned*)(kh + RHN);
  int* qpos = (int*)(mm + 128);
  int* kpos = qpos + RHN;

  hept_init<<<1, 64, 0, stream>>>(mm);
  hept_hash<<<dim3(PADDED / 256, N_HEADS, N_HASHES), 256, 0, stream>>>(
      query, key, alpha, qh, kh, mm);

  // 256KB dynamic LDS for the in-LDS bitonic sort (320KB available per WGP)
  (void)hipFuncSetAttribute((const void*)hept_sort,
                            hipFuncAttributeMaxDynamicSharedMemorySize,
                            PADDED * 8);
  hept_sort<<<2 * N_HASHES * N_HEADS, 1024, PADDED * 8, stream>>>(
      qh, kh, shifts, mm, qpos, kpos);

  hept_attn<<<dim3(NB, N_HEADS, N_HASHES), 256, 0, stream>>>(
      query, key, value, qpos, kpos, out);
}